// GATLayer_34291018891894
// MI455X (gfx1250) — compile-verified
//
#include <hip/hip_runtime.h>

// ---------------------------------------------------------------------------
// GAT layer, fused flash-softmax formulation for MI455X (gfx1250, wave32).
//   ltg = graph @ W                          (WMMA f32 16x16x4)
//   s_i = ltg_i . a[:64], t_j = ltg_j . a[64:]
//   m_i = lrelu(s_i + max_j t_j)             (exact row max, lrelu monotone)
//   P_ij = exp(lrelu(s_i + t_j) - m_i)       (computed on the fly, never stored)
//   H_i  = (sum_j P_ij * ltg_j) / (sum_j P_ij)   (WMMA f32 16x16x4, K streamed)
// ---------------------------------------------------------------------------

typedef __attribute__((ext_vector_type(2))) float v2f;
typedef __attribute__((ext_vector_type(8))) float v8f;

#define N_NODES 8192
#define FDIM    64
#define TILES   (N_NODES / 16)   // 512 row tiles
#define SPLIT   8                // K-dimension split -> 4096 waves in attn

__device__ __forceinline__ float lrelu(float x) { return x > 0.f ? x : 0.2f * x; }

// order-preserving float -> uint key (so memset-0 is a valid "-inf" init)
__device__ __forceinline__ unsigned fenc(float x) {
    unsigned b = __float_as_uint(x);
    return (b & 0x80000000u) ? ~b : (b | 0x80000000u);
}
__device__ __forceinline__ float fdec(unsigned k) {
    return (k & 0x80000000u) ? __uint_as_float(k & 0x7FFFFFFFu) : __uint_as_float(~k);
}

// ---------------------------------------------------------------------------
// Kernel 1: ltg = graph @ W.  One wave per 16-row tile. K=64 in chunks of 4.
// A layout (16x4 f32): lanes 0-15 = rows, VGPR0 = K {0 | 2}, VGPR1 = K {1 | 3}
// B layout (4x16 f32): lanes = cols, VGPR0 = K {0 | 2}, VGPR1 = K {1 | 3}
// C/D (16x16 f32): VGPR v, lane L -> row v + 8*(L>=16), col L&15
// ---------------------------------------------------------------------------
__global__ __launch_bounds__(32) void k_gemm_ltg(const float* __restrict__ g,
                                                 const float* __restrict__ W,
                                                 float* __restrict__ ltg) {
    const int i0   = blockIdx.x * 16;
    const int lane = threadIdx.x;
    const int half = lane >> 4;     // 0 | 1
    const int l    = lane & 15;

    v8f acc0 = {}, acc1 = {}, acc2 = {}, acc3 = {};

    for (int k0 = 0; k0 < FDIM; k0 += 4) {
        const int ka = k0 + half * 2;
        v2f a;                                      // graph[i0+l][ka], [ka+1]
        const float* gp = g + (i0 + l) * FDIM + ka;
        a.x = gp[0]; a.y = gp[1];

        const float* w0 = W + ka * FDIM;            // row ka
        const float* w1 = W + (ka + 1) * FDIM;      // row ka+1
        v2f b0, b1, b2, b3;
        b0.x = w0[l];      b0.y = w1[l];
        b1.x = w0[16 + l]; b1.y = w1[16 + l];
        b2.x = w0[32 + l]; b2.y = w1[32 + l];
        b3.x = w0[48 + l]; b3.y = w1[48 + l];

        acc0 = __builtin_amdgcn_wmma_f32_16x16x4_f32(false, a, false, b0, (short)0, acc0, false, false);
        acc1 = __builtin_amdgcn_wmma_f32_16x16x4_f32(false, a, false, b1, (short)0, acc1, false, false);
        acc2 = __builtin_amdgcn_wmma_f32_16x16x4_f32(false, a, false, b2, (short)0, acc2, false, false);
        acc3 = __builtin_amdgcn_wmma_f32_16x16x4_f32(false, a, false, b3, (short)0, acc3, false, false);
    }

    const int rbase = i0 + half * 8;
#pragma unroll
    for (int v = 0; v < 8; ++v) {
        float* op = ltg + (rbase + v) * FDIM + l;
        op[0]  = acc0[v];
        op[16] = acc1[v];
        op[32] = acc2[v];
        op[48] = acc3[v];
    }
}

// ---------------------------------------------------------------------------
// Kernel 2: per-row s, t and global max of t (atomic on encoded uint key).
// ---------------------------------------------------------------------------
__global__ void k_row_sv(const float* __restrict__ ltg, const float* __restrict__ av,
                         float* __restrict__ S, float* __restrict__ T,
                         unsigned* __restrict__ tmaxk) {
    const int i = blockIdx.x * blockDim.x + threadIdx.x;
    if (i >= N_NODES) return;
    const float* r = ltg + i * FDIM;
    float s = 0.f, t = 0.f;
#pragma unroll 8
    for (int f = 0; f < FDIM; ++f) {
        const float v = r[f];
        s += v * av[f];
        t += v * av[FDIM + f];
    }
    S[i] = s;
    T[i] = t;
    atomicMax(tmaxk, fenc(t));
}

// ---------------------------------------------------------------------------
// Kernel 3: fused exp-weighted accumulation.  One wave per (row-tile, j-slice).
// A fragment = softmax weights computed in-register; B fragment = ltg rows.
// Partial numerators/denominators combined with global f32 atomics.
// ---------------------------------------------------------------------------
__global__ __launch_bounds__(32) void k_attn(const float* __restrict__ ltg,
                                             const float* __restrict__ S,
                                             const float* __restrict__ T,
                                             const unsigned* __restrict__ tmaxk,
                                             float* __restrict__ Hnum,
                                             float* __restrict__ Z) {
    const int tile  = blockIdx.x / SPLIT;
    const int slice = blockIdx.x % SPLIT;
    const int i0    = tile * 16;
    const int lane  = threadIdx.x;
    const int half  = lane >> 4;
    const int l     = lane & 15;

    const float srow = S[i0 + l];
    const float m    = lrelu(srow + fdec(*tmaxk));   // exact row max of E

    const int JPS  = N_NODES / SPLIT;
    const int jbeg = slice * JPS;

    v8f acc0 = {}, acc1 = {}, acc2 = {}, acc3 = {};
    float z = 0.f;

#pragma unroll 2
    for (int j0 = jbeg; j0 < jbeg + JPS; j0 += 4) {
        const int ja = j0 + half * 2;
        const v2f tt = *(const v2f*)(T + ja);        // t[ja], t[ja+1]

        v2f a;                                        // softmax weights (A frag)
        a.x = __expf(lrelu(srow + tt.x) - m);
        a.y = __expf(lrelu(srow + tt.y) - m);
        z += a.x + a.y;

        const float* r0 = ltg + ja * FDIM;            // ltg row ja
        const float* r1 = r0 + FDIM;                  // ltg row ja+1
        v2f b0, b1, b2, b3;
        b0.x = r0[l];      b0.y = r1[l];
        b1.x = r0[16 + l]; b1.y = r1[16 + l];
        b2.x = r0[32 + l]; b2.y = r1[32 + l];
        b3.x = r0[48 + l]; b3.y = r1[48 + l];

        acc0 = __builtin_amdgcn_wmma_f32_16x16x4_f32(false, a, false, b0, (short)0, acc0, false, false);
        acc1 = __builtin_amdgcn_wmma_f32_16x16x4_f32(false, a, false, b1, (short)0, acc1, false, false);
        acc2 = __builtin_amdgcn_wmma_f32_16x16x4_f32(false, a, false, b2, (short)0, acc2, false, false);
        acc3 = __builtin_amdgcn_wmma_f32_16x16x4_f32(false, a, false, b3, (short)0, acc3, false, false);
    }

    // denominator: lane L covers j%4 in {0,1}, lane L+16 covers {2,3} -> both add
    atomicAdd(&Z[i0 + l], z);

    const int rbase = i0 + half * 8;
#pragma unroll
    for (int v = 0; v < 8; ++v) {
        float* hp = Hnum + (rbase + v) * FDIM + l;
        atomicAdd(hp + 0,  acc0[v]);
        atomicAdd(hp + 16, acc1[v]);
        atomicAdd(hp + 32, acc2[v]);
        atomicAdd(hp + 48, acc3[v]);
    }
}

// ---------------------------------------------------------------------------
// Kernel 4: H = Hnum / Z  (overwrites ltg staging in d_out)
// ---------------------------------------------------------------------------
__global__ void k_fin(const float* __restrict__ Hnum, const float* __restrict__ Z,
                      float* __restrict__ out) {
    const int idx = blockIdx.x * blockDim.x + threadIdx.x;
    if (idx >= N_NODES * FDIM) return;
    out[idx] = Hnum[idx] / Z[idx >> 6];
}

extern "C" void kernel_launch(void* const* d_in, const int* in_sizes, int n_in,
                              void* d_out, int out_size, void* d_ws, size_t ws_size,
                              hipStream_t stream) {
    const float* g  = (const float*)d_in[0];   // [8192, 64]
    const float* W  = (const float*)d_in[1];   // [64, 64]
    const float* av = (const float*)d_in[2];   // [128, 1]
    float* out = (float*)d_out;                // [8192, 64] (also stages ltg)
    float* ws  = (float*)d_ws;

    // workspace layout (floats): Hnum[524288] | Z[8192] | tmax-key + pad [8] | S[8192] | T[8192]
    float*    Hnum  = ws;
    float*    Z     = ws + (size_t)N_NODES * FDIM;
    unsigned* tmaxk = (unsigned*)(Z + N_NODES);
    float*    S     = Z + N_NODES + 8;
    float*    T     = S + N_NODES;

    // zero Hnum, Z and tmax key (key 0 == below any real float's key)
    hipMemsetAsync(ws, 0, ((size_t)N_NODES * FDIM + N_NODES + 8) * sizeof(float), stream);

    k_gemm_ltg<<<TILES, 32, 0, stream>>>(g, W, out);                    // ltg -> d_out
    k_row_sv<<<N_NODES / 256, 256, 0, stream>>>(out, av, S, T, tmaxk);
    k_attn<<<TILES * SPLIT, 32, 0, stream>>>(out, S, T, tmaxk, Hnum, Z);
    k_fin<<<(N_NODES * FDIM) / 256, 256, 0, stream>>>(Hnum, Z, out);    // H -> d_out
}